// HyperMixer_9637906612574
// MI455X (gfx1250) — compile-verified
//
#include <hip/hip_runtime.h>
#include <math.h>

#define EPS 1e-6f
#define KSTREAMS 4
#define DHID 2048
#define KD 8192            // KSTREAMS * DHID
#define OUTDIM 24          // (2 + K) * K
#define OUTPAD 32          // OUTDIM padded to 2 WMMA N-tiles
#define NTOK 8192          // B * S
#define SINKHORN_ITERS 20

typedef float v2f __attribute__((ext_vector_type(2)));
typedef float v8f __attribute__((ext_vector_type(8)));

// ---------------------------------------------------------------------------
// Phase 0: zero-pad W [24][8192] -> Wp [32][8192] so the WMMA K-loop is
// completely branch-free (EXEC all-ones, unconditional b64 loads).
// ---------------------------------------------------------------------------
__global__ __launch_bounds__(256) void hm_padW(const float* __restrict__ W,
                                               float* __restrict__ Wp) {
  const size_t i = (size_t)blockIdx.x * blockDim.x + threadIdx.x;  // 0 .. 32*8192-1
  const size_t row = i / KD;
  Wp[i] = (row < OUTDIM) ? W[i] : 0.0f;   // rows 0..23 copied, 24..31 zero
}

// ---------------------------------------------------------------------------
// Phase 1: per-token RMS stats + logits projection via V_WMMA_F32_16X16X4_F32.
// logits[t][o] = rsqrt(mean(x_t^2)+eps) * dot(x_t, W[o])   (rsqrt folded out
// of the GEMM since it is a per-token scalar).
// One wave32 per 16-token tile; 2 WMMA per K-step (N tiles 0-15 and 16-31,
// where N>=24 is zero padding).
//
// fp32 WMMA operand layout (ISA 7.12.2):
//   A 16x4:  lane = M + 16*(K>=2), VGPR0 = K={0|2}, VGPR1 = K={1|3}
//   B 4x16:  lane = N + 16*(K>=2), VGPR0 = K={0|2}, VGPR1 = K={1|3}
//   D 16x16: VGPR i: lanes 0-15 -> M=i, N=lane; lanes 16-31 -> M=i+8, N=lane-16
// ---------------------------------------------------------------------------
__global__ __launch_bounds__(32) void hm_logits(const float* __restrict__ x,
                                                const float* __restrict__ Wp,
                                                float* __restrict__ logits) {
  const int lane = threadIdx.x;          // wave32, all lanes active (WMMA req)
  const int t0   = blockIdx.x * 16;      // 16 tokens per wave
  const int m    = lane & 15;
  const int kk   = (lane >> 4) * 2;      // lanes 0-15: K={0,1}; 16-31: K={2,3}
  const int n1   = 16 + m;               // second N tile (>=24 is zero pad)

  const float* xrow = x  + (size_t)(t0 + m) * KD + kk;
  const float* wp0  = Wp + (size_t)m  * KD + kk;   // rows  0..15
  const float* wp1  = Wp + (size_t)n1 * KD + kk;   // rows 16..31 (padded)

  v8f c0 = {0.f, 0.f, 0.f, 0.f, 0.f, 0.f, 0.f, 0.f};
  v8f c1 = {0.f, 0.f, 0.f, 0.f, 0.f, 0.f, 0.f, 0.f};
  float sumsq = 0.f;

  for (int k = 0; k < KD; k += 4) {
    const v2f a  = *(const v2f*)(xrow + k);
    const v2f b0 = *(const v2f*)(wp0 + k);
    const v2f b1 = *(const v2f*)(wp1 + k);
    sumsq = fmaf(a.x, a.x, sumsq);
    sumsq = fmaf(a.y, a.y, sumsq);
    c0 = __builtin_amdgcn_wmma_f32_16x16x4_f32(false, a, false, b0,
                                               (short)0, c0, false, false);
    c1 = __builtin_amdgcn_wmma_f32_16x16x4_f32(false, a, false, b1,
                                               (short)0, c1, false, false);
  }

  // lane l holds K%4 in {kk,kk+1} partial sumsq of token (l&15); fold halves.
  sumsq += __shfl_xor(sumsq, 16);
  const float rs = rsqrtf(sumsq * (1.0f / (float)KD) + EPS);

  const int half = (lane >> 4) * 8;
  #pragma unroll
  for (int i = 0; i < 8; ++i) {
    const int   mrow = i + half;
    const float r    = __shfl(rs, mrow);  // rsqrt of token t0+mrow is in lane mrow
    float* dst = logits + (size_t)(t0 + mrow) * OUTPAD;  // 32-float padded rows
    dst[m]  = c0[i] * r;
    dst[n1] = c1[i] * r;                  // cols 24..31 are dead padding
  }
}

// ---------------------------------------------------------------------------
// Phase 2: per-token gating weights (sigmoids, 4x4 softmax, Sinkhorn).
// One thread per token; everything lives in registers.
// ---------------------------------------------------------------------------
__global__ __launch_bounds__(256) void hm_weights(const float* __restrict__ logits,
                                                  const float* __restrict__ scale,
                                                  const float* __restrict__ base,
                                                  float* __restrict__ wts) {
  const int t = blockIdx.x * blockDim.x + threadIdx.x;
  if (t >= NTOK) return;
  const float* lg = logits + (size_t)t * OUTPAD;
  const float s0 = scale[0], s1 = scale[1], s2 = scale[2];

  float o[OUTDIM];
  #pragma unroll
  for (int k = 0; k < 4; ++k)
    o[k] = 1.0f / (1.0f + expf(-(lg[k] * s0 + base[k]))) + EPS;       // pre
  #pragma unroll
  for (int k = 0; k < 4; ++k)
    o[4 + k] = 2.0f / (1.0f + expf(-(lg[4 + k] * s1 + base[4 + k]))); // post

  float m[4][4];
  #pragma unroll
  for (int h = 0; h < 4; ++h) {
    float row[4], mx = -INFINITY;
    #pragma unroll
    for (int k = 0; k < 4; ++k) {
      row[k] = lg[8 + h * 4 + k] * s2 + base[8 + h * 4 + k];
      mx = fmaxf(mx, row[k]);
    }
    float s = 0.f;
    #pragma unroll
    for (int k = 0; k < 4; ++k) { row[k] = expf(row[k] - mx); s += row[k]; }
    const float inv = 1.0f / s;
    #pragma unroll
    for (int k = 0; k < 4; ++k) m[h][k] = row[k] * inv + EPS;
  }

  // Sinkhorn: priming column norm, then 19 x (row norm, column norm)
  #pragma unroll
  for (int k = 0; k < 4; ++k) {
    float cs = m[0][k] + m[1][k] + m[2][k] + m[3][k];
    float inv = 1.0f / (cs + EPS);
    #pragma unroll
    for (int h = 0; h < 4; ++h) m[h][k] *= inv;
  }
  for (int it = 0; it < SINKHORN_ITERS - 1; ++it) {
    #pragma unroll
    for (int h = 0; h < 4; ++h) {
      float rsum = m[h][0] + m[h][1] + m[h][2] + m[h][3];
      float inv = 1.0f / (rsum + EPS);
      #pragma unroll
      for (int k = 0; k < 4; ++k) m[h][k] *= inv;
    }
    #pragma unroll
    for (int k = 0; k < 4; ++k) {
      float cs = m[0][k] + m[1][k] + m[2][k] + m[3][k];
      float inv = 1.0f / (cs + EPS);
      #pragma unroll
      for (int h = 0; h < 4; ++h) m[h][k] *= inv;
    }
  }
  #pragma unroll
  for (int h = 0; h < 4; ++h)
    #pragma unroll
    for (int k = 0; k < 4; ++k) o[8 + h * 4 + k] = m[h][k];

  float* w = wts + (size_t)t * OUTDIM;
  #pragma unroll
  for (int j = 0; j < OUTDIM; ++j) w[j] = o[j];
}

// ---------------------------------------------------------------------------
// Phase 3: bandwidth-bound collapse + expand, float4-vectorized.
// One block per token; 256 threads x 2 float4 each cover D=2048.
// ---------------------------------------------------------------------------
__global__ __launch_bounds__(256) void hm_mix(const float* __restrict__ x,
                                              const float* __restrict__ outp,
                                              const float* __restrict__ wts,
                                              float* __restrict__ y,
                                              float* __restrict__ coll) {
  const int t = blockIdx.x;
  __shared__ float w[OUTDIM];
  if (threadIdx.x < OUTDIM) w[threadIdx.x] = wts[(size_t)t * OUTDIM + threadIdx.x];
  __syncthreads();

  const float pre0 = w[0], pre1 = w[1], pre2 = w[2], pre3 = w[3];
  const float* xb = x    + (size_t)t * (KSTREAMS * DHID);
  const float* ob = outp + (size_t)t * DHID;
  float*       yb = y    + (size_t)t * (KSTREAMS * DHID);
  float*       cb = coll + (size_t)t * DHID;

  for (int d = threadIdx.x * 4; d < DHID; d += 256 * 4) {
    const float4 x0 = *(const float4*)(xb + 0 * DHID + d);
    const float4 x1 = *(const float4*)(xb + 1 * DHID + d);
    const float4 x2 = *(const float4*)(xb + 2 * DHID + d);
    const float4 x3 = *(const float4*)(xb + 3 * DHID + d);
    const float4 o  = *(const float4*)(ob + d);

    float4 c;
    c.x = pre0 * x0.x + pre1 * x1.x + pre2 * x2.x + pre3 * x3.x;
    c.y = pre0 * x0.y + pre1 * x1.y + pre2 * x2.y + pre3 * x3.y;
    c.z = pre0 * x0.z + pre1 * x1.z + pre2 * x2.z + pre3 * x3.z;
    c.w = pre0 * x0.w + pre1 * x1.w + pre2 * x2.w + pre3 * x3.w;
    *(float4*)(cb + d) = c;

    #pragma unroll
    for (int h = 0; h < 4; ++h) {
      const float ph = w[4 + h];
      const float c0 = w[8 + h * 4 + 0];
      const float c1 = w[8 + h * 4 + 1];
      const float c2 = w[8 + h * 4 + 2];
      const float c3 = w[8 + h * 4 + 3];
      float4 r;
      r.x = ph * o.x + c0 * x0.x + c1 * x1.x + c2 * x2.x + c3 * x3.x;
      r.y = ph * o.y + c0 * x0.y + c1 * x1.y + c2 * x2.y + c3 * x3.y;
      r.z = ph * o.z + c0 * x0.z + c1 * x1.z + c2 * x2.z + c3 * x3.z;
      r.w = ph * o.w + c0 * x0.w + c1 * x1.w + c2 * x2.w + c3 * x3.w;
      *(float4*)(yb + h * DHID + d) = r;
    }
  }
}

// ---------------------------------------------------------------------------
extern "C" void kernel_launch(void* const* d_in, const int* in_sizes, int n_in,
                              void* d_out, int out_size, void* d_ws, size_t ws_size,
                              hipStream_t stream) {
  const float* x     = (const float*)d_in[0];  // [2,4096,4,2048]
  const float* outp  = (const float*)d_in[1];  // [2,4096,2048]
  const float* W     = (const float*)d_in[2];  // [24,8192]
  const float* scale = (const float*)d_in[3];  // [3]
  const float* base  = (const float*)d_in[4];  // [24]

  float* y    = (float*)d_out;                          // [2,4096,4,2048]
  float* coll = y + (size_t)NTOK * KSTREAMS * DHID;     // [2,4096,2048]

  float* Wp     = (float*)d_ws;                       // 32*8192 floats (1 MB)
  float* logits = Wp + (size_t)OUTPAD * KD;           // NTOK*32 floats (1 MB)
  float* wts    = logits + (size_t)NTOK * OUTPAD;     // NTOK*24 floats (768 KB)

  hm_padW   <<<(OUTPAD * KD) / 256, 256, 0, stream>>>(W, Wp);
  hm_logits <<<NTOK / 16, 32, 0, stream>>>(x, Wp, logits);
  hm_weights<<<(NTOK + 255) / 256, 256, 0, stream>>>(logits, scale, base, wts);
  hm_mix    <<<NTOK, 256, 0, stream>>>(x, outp, wts, y, coll);
}